// ARLayer_56298431316389
// MI455X (gfx1250) — compile-verified
//
#include <hip/hip_runtime.h>
#include <math.h>

// Problem constants (from reference): B=64, S=2048, D=512, 4 context vectors.
#define B_  64
#define S_  2048
#define D_  512
#define NC_ 4
#define NCHUNK 8                 // s-chunks per batch row -> 8*64 = 512 blocks
#define SCHUNK (S_ / NCHUNK)     // 256 sequence rows per block
#define UPAD   (D_ + 4)          // LDS row stride: 516 % 64 == 4 -> conflict-free b64 reads

typedef float v2f __attribute__((ext_vector_type(2)));
typedef float v8f __attribute__((ext_vector_type(8)));

// ---------------------------------------------------------------------------
// Kernel A: U[c][b][d] = bias[d] + sum_e kernel[d][e] * cvec_c[b][e]
// (folds bias so that scores[c,b,s] = sent[b,s,:] . U[c,b,:])
// ---------------------------------------------------------------------------
__global__ void __launch_bounds__(256)
make_u_kernel(const float* __restrict__ c1, const float* __restrict__ c2,
              const float* __restrict__ c3, const float* __restrict__ c4,
              const float* __restrict__ Kmat, const float* __restrict__ bias,
              float* __restrict__ U) {
  const int c = blockIdx.x;   // 0..3
  const int b = blockIdx.y;   // 0..63
  const float* cvec = (c == 0 ? c1 : c == 1 ? c2 : c == 2 ? c3 : c4) + (size_t)b * D_;

  __shared__ float lds_c[D_];
  for (int i = threadIdx.x; i < D_; i += blockDim.x) lds_c[i] = cvec[i];
  __syncthreads();

  const int wave = threadIdx.x >> 5;
  const int lane = threadIdx.x & 31;
  for (int d = wave * 64; d < wave * 64 + 64; ++d) {
    const float* krow = Kmat + (size_t)d * D_;
    float acc = 0.f;
    for (int e = lane; e < D_; e += 32) acc += krow[e] * lds_c[e];  // coalesced
    for (int m = 16; m > 0; m >>= 1) acc += __shfl_xor(acc, m, 32); // wave32 reduce
    if (lane == 0) U[((size_t)c * B_ + b) * D_ + d] = acc + bias[d];
  }
}

// ---------------------------------------------------------------------------
// Kernel B: fused single-read pass over sent. Block = (chunk, b), 8 wave32.
//   phase 1: scores via chained V_WMMA_F32_16X16X4_F32; B matrix is u^T
//            zero-padded to 16 columns in LDS -> branchless ds_load_b64 frags
//   phase 2: local softmax partials (m, l) + attn weights in LDS
//   phase 3: P_partial[4][512] = sum_s attn * sent[s,:]  (tile is L2-hot)
// ---------------------------------------------------------------------------
__global__ void __launch_bounds__(256)
attn_pass_kernel(const float* __restrict__ sent, const float* __restrict__ U,
                 float* __restrict__ pM, float* __restrict__ pL,
                 float* __restrict__ pP) {
  const int chunk = blockIdx.x;
  const int b     = blockIdx.y;
  const int tid   = threadIdx.x;
  const int wave  = tid >> 5;
  const int lane  = tid & 31;
  const int col   = lane & 15;   // WMMA N index / A-row M this lane feeds
  const int khalf = lane >> 4;   // 0: K={0,1}, 1: K={2,3} (ISA 16x4 f32 layout)

  __shared__ float lds_u[16][UPAD];      // ~33 KB, rows 4..15 zero (B cols 4..15)
  __shared__ float lds_sc[16][SCHUNK];   // 16 KB, rows 0..3 meaningful

  // zero-pad then fill the 4 real u rows (bias already folded in U)
  for (int i = tid; i < 16 * UPAD; i += 256) (&lds_u[0][0])[i] = 0.f;
  __syncthreads();
  for (int i = tid; i < NC_ * D_; i += 256)
    lds_u[i / D_][i % D_] = U[((size_t)(i / D_) * B_ + b) * D_ + (i % D_)];
  __syncthreads();

  const int s0 = chunk * SCHUNK;
  const float* sb = sent + ((size_t)b * S_ + s0) * D_;

  // ---- phase 1: scores via WMMA (2 tiles of 16 rows per wave) -------------
  for (int t = 0; t < SCHUNK / 128; ++t) {
    const int rloc = t * 128 + wave * 16 + col;              // local s-row (tile M)
    const v2f* ag = (const v2f*)(sb + (size_t)rloc * D_);    // 8B-aligned row
    v8f acc = {};
#pragma unroll 8
    for (int kk = 0; kk < D_ / 4; ++kk) {                    // k0 = 4*kk
      v2f a  = ag[2 * kk + khalf];                           // A: (M=col, K pair)
      v2f bf = *(const v2f*)&lds_u[col][4 * kk + 2 * khalf]; // B: (K pair, N=col)
      acc = __builtin_amdgcn_wmma_f32_16x16x4_f32(false, a, false, bf,
                                                  (short)0, acc, false, false);
    }
    // C/D layout: VGPR v holds (M = v + 8*khalf, N = col); rows 4..15 unused
#pragma unroll
    for (int v = 0; v < 8; ++v)
      lds_sc[col][t * 128 + wave * 16 + v + 8 * khalf] = acc[v];
  }
  __syncthreads();

  // ---- phase 2: local softmax partials (waves 0..3, one per c) ------------
  if (wave < NC_) {
    const int c = wave;
    float m = -__builtin_inff();
    for (int i = lane; i < SCHUNK; i += 32) m = fmaxf(m, lds_sc[c][i]);
    for (int s = 16; s > 0; s >>= 1) m = fmaxf(m, __shfl_xor(m, s, 32));
    float l = 0.f;
    for (int i = lane; i < SCHUNK; i += 32) {
      float e = __expf(lds_sc[c][i] - m);
      lds_sc[c][i] = e;                     // scores -> attn weights in place
      l += e;
    }
    for (int s = 16; s > 0; s >>= 1) l += __shfl_xor(l, s, 32);
    if (lane == 0) {
      pM[((size_t)chunk * B_ + b) * NC_ + c] = m;
      pL[((size_t)chunk * B_ + b) * NC_ + c] = l;
    }
  }
  __syncthreads();

  // ---- phase 3: P_partial = attn^T * sent_tile (coalesced float2 stream) --
  const float2* sb2 = (const float2*)sb;    // thread t owns d = 2t, 2t+1
  float2 acc[NC_];
#pragma unroll
  for (int c = 0; c < NC_; ++c) acc[c] = make_float2(0.f, 0.f);
#pragma unroll 4
  for (int s = 0; s < SCHUNK; ++s) {
    float2 v = sb2[(size_t)s * (D_ / 2) + tid];
#pragma unroll
    for (int c = 0; c < NC_; ++c) {
      float w = lds_sc[c][s];               // LDS broadcast
      acc[c].x += w * v.x;
      acc[c].y += w * v.y;
    }
  }
  float2* pP2 = (float2*)pP;
#pragma unroll
  for (int c = 0; c < NC_; ++c)
    pP2[(((size_t)chunk * B_ + b) * NC_ + c) * (D_ / 2) + tid] = acc[c];
}

// ---------------------------------------------------------------------------
// Kernel C: combine chunk partials with softmax rescaling, write P[4][B][D].
// out = (sum_j e^{m_j - M} * P_j) / (sum_j e^{m_j - M} * l_j)
// ---------------------------------------------------------------------------
__global__ void __launch_bounds__(256)
finalize_kernel(const float* __restrict__ pM, const float* __restrict__ pL,
                const float* __restrict__ pP, float* __restrict__ out) {
  const int c = blockIdx.x;
  const int b = blockIdx.y;
  float m[NCHUNK], l[NCHUNK], w[NCHUNK];
  float M = -__builtin_inff();
#pragma unroll
  for (int j = 0; j < NCHUNK; ++j) {
    m[j] = pM[((size_t)j * B_ + b) * NC_ + c];
    l[j] = pL[((size_t)j * B_ + b) * NC_ + c];
    M = fmaxf(M, m[j]);
  }
  float L = 0.f;
#pragma unroll
  for (int j = 0; j < NCHUNK; ++j) { w[j] = __expf(m[j] - M); L += w[j] * l[j]; }
  const float inv = 1.f / L;
  for (int d = threadIdx.x; d < D_; d += blockDim.x) {
    float acc = 0.f;
#pragma unroll
    for (int j = 0; j < NCHUNK; ++j)
      acc += w[j] * pP[(((size_t)j * B_ + b) * NC_ + c) * D_ + d];
    out[((size_t)c * B_ + b) * D_ + d] = acc * inv;
  }
}

// ---------------------------------------------------------------------------
extern "C" void kernel_launch(void* const* d_in, const int* in_sizes, int n_in,
                              void* d_out, int out_size, void* d_ws, size_t ws_size,
                              hipStream_t stream) {
  const float* sent = (const float*)d_in[0];   // [B,S,D]
  const float* c1   = (const float*)d_in[1];   // [B,D]
  const float* c2   = (const float*)d_in[2];
  const float* c3   = (const float*)d_in[3];
  const float* c4   = (const float*)d_in[4];
  const float* Kmat = (const float*)d_in[5];   // [D,D]
  const float* bias = (const float*)d_in[6];   // [D,1]

  // Workspace layout (floats): U | pM | pL | pP  (~4.5 MB total)
  float* U  = (float*)d_ws;                                  // 4*64*512
  float* pM = U  + (size_t)NC_ * B_ * D_;                    // 8*64*4
  float* pL = pM + (size_t)NCHUNK * B_ * NC_;                // 8*64*4
  float* pP = pL + (size_t)NCHUNK * B_ * NC_;                // 8*64*4*512

  make_u_kernel<<<dim3(NC_, B_), 256, 0, stream>>>(c1, c2, c3, c4, Kmat, bias, U);
  attn_pass_kernel<<<dim3(NCHUNK, B_), 256, 0, stream>>>(sent, U, pM, pL, pP);
  finalize_kernel<<<dim3(NC_, B_), 256, 0, stream>>>(pM, pL, pP, (float*)d_out);
}